// PointFeatureAlignment_17506286698529
// MI455X (gfx1250) — compile-verified
//
#include <hip/hip_runtime.h>
#include <math.h>

typedef __attribute__((ext_vector_type(2))) float v2f;
typedef __attribute__((ext_vector_type(8))) float v8f;

#define IMG  224
#define NPIX (IMG * IMG)
#define MAXWIN 256

// ---------------------------------------------------------------------------
// Kernel 1: one wave (32 lanes) per 16 target points.
// score(m,s) = |src_s|^2 - 2 * tgt_m . src_s computed via V_WMMA_F32_16X16X4_F32
// (A row = (tx,ty,tz,1), B col = (-2sx,-2sy,-2sz,|s|^2), C = 0). The per-row
// |t|^2 constant does not change top-3 ordering and is added back afterwards.
// The 16 x G score block is spilled to LDS; all 32 lanes then cooperatively
// scan (two half-row scans + sorted-triple merge) for top-3 and emit
// (idx[3], weight[3]) per target point.
// ---------------------------------------------------------------------------
__global__ __launch_bounds__(32)
void knn_weights_kernel(const float* __restrict__ tgt,   // (B,N,3)
                        const float* __restrict__ src,   // (B,G,3)
                        int*   __restrict__ idx3,        // (B,N,3)
                        float* __restrict__ w3,          // (B,N,3)
                        int B, int N, int G)
{
    extern __shared__ float smem[];
    float* scores = smem;            // 16 * G floats
    float* tsq_sh = smem + 16 * G;   // 16 floats

    const int lane     = (int)threadIdx.x;      // 0..31
    const int tilesPer = (N + 15) >> 4;
    const int tile     = (int)blockIdx.x;
    const int b        = tile / tilesPer;
    const int m0       = (tile - b * tilesPer) << 4;

    const int  col = lane & 15;
    const bool hi  = lane >= 16;
    const int  m   = m0 + col;
    const int  mc  = m < N ? m : (N - 1);       // clamp tail reads

    const float* tp = tgt + ((size_t)b * N + mc) * 3;
    const float tx = tp[0], ty = tp[1], tz = tp[2];
    if (!hi) tsq_sh[col] = tx * tx + ty * ty + tz * tz;

    // A-matrix (16x4 f32): lanes 0-15 hold K=0,1 ; lanes 16-31 hold K=2,3
    v2f a;
    a.x = hi ? tz   : tx;
    a.y = hi ? 1.0f : ty;

    const float* sb     = src + (size_t)b * G * 3;
    const int    chunks = G >> 4;
    const int    rbase  = hi ? 8 : 0;
    float* stbase = scores + rbase * G + col;   // per-lane LDS store base

    for (int c = 0; c < chunks; ++c) {
        const int s = (c << 4) + col;           // source column this lane owns
        const float* sp = sb + s * 3;
        const float sx = sp[0], sy = sp[1], sz = sp[2];
        const float ssq   = sx * sx + sy * sy + sz * sz;
        const float bx_lo = -2.0f * sx, by_lo = -2.0f * sy;
        const float bx_hi = -2.0f * sz;

        // B-matrix (4x16 f32): lanes 0-15 hold rows K=0,1 ; lanes 16-31 rows K=2,3
        v2f bm;
        bm.x = hi ? bx_hi : bx_lo;              // branchless -> v_cndmask
        bm.y = hi ? ssq   : by_lo;

        v8f cz = {};
        v8f d = __builtin_amdgcn_wmma_f32_16x16x4_f32(
            /*neg_a=*/false, a, /*neg_b=*/false, bm,
            /*c_mod=*/(short)0, cz, /*reuse_a=*/false, /*reuse_b=*/false);

        // C/D layout: element r in lane l -> row r + (l>=16 ? 8 : 0), col l%16
        #pragma unroll
        for (int r = 0; r < 8; ++r)
            stbase[r * G + (c << 4)] = d[r];
    }
    __syncthreads();

    // Cooperative top-3: lane l scans half (l>>4) of row (l&15).
    const float* srow = scores + col * G;
    const int half  = G >> 1;
    const int sbeg  = hi ? half : 0;
    const int send  = sbeg + half;
    float s0 = 3.4e38f, s1 = 3.4e38f, s2 = 3.4e38f;
    int   i0 = 0,       i1 = 0,       i2 = 0;
    for (int s = sbeg; s < send; ++s) {
        const float v = srow[s];
        if      (v < s0) { s2 = s1; i2 = i1; s1 = s0; i1 = i0; s0 = v; i0 = s; }
        else if (v < s1) { s2 = s1; i2 = i1; s1 = v;  i1 = s; }
        else if (v < s2) { s2 = v;  i2 = s; }
    }
    __syncthreads();                             // done reading scores; reuse LDS

    float* mS = scores;                          // 32*3 floats
    int*   mI = (int*)(scores + 32 * 3);         // 32*3 ints
    mS[lane * 3 + 0] = s0; mS[lane * 3 + 1] = s1; mS[lane * 3 + 2] = s2;
    mI[lane * 3 + 0] = i0; mI[lane * 3 + 1] = i1; mI[lane * 3 + 2] = i2;
    __syncthreads();

    if (!hi && m < N) {
        // Merge two sorted triples; ties prefer the low-column half (<=),
        // matching lax.top_k's lower-index tie-breaking.
        float av[3], bv[3], ov[3];
        int   aI[3], bI[3], oI[3];
        #pragma unroll
        for (int k = 0; k < 3; ++k) {
            av[k] = mS[lane * 3 + k];        aI[k] = mI[lane * 3 + k];
            bv[k] = mS[(lane + 16) * 3 + k]; bI[k] = mI[(lane + 16) * 3 + k];
        }
        int ai = 0, bi = 0;
        #pragma unroll
        for (int k = 0; k < 3; ++k) {
            if (av[ai] <= bv[bi]) { ov[k] = av[ai]; oI[k] = aI[ai]; ++ai; }
            else                  { ov[k] = bv[bi]; oI[k] = bI[bi]; ++bi; }
        }
        const float tq = tsq_sh[lane];
        const float d0 = fmaxf(ov[0] + tq, 1e-10f);
        const float d1 = fmaxf(ov[1] + tq, 1e-10f);
        const float d2 = fmaxf(ov[2] + tq, 1e-10f);
        const float r0 = 1.0f / d0, r1 = 1.0f / d1, r2 = 1.0f / d2;
        const float inv = 1.0f / (r0 + r1 + r2);
        const size_t o = ((size_t)b * N + m) * 3;
        idx3[o + 0] = oI[0];    idx3[o + 1] = oI[1];    idx3[o + 2] = oI[2];
        w3  [o + 0] = r0 * inv; w3  [o + 1] = r1 * inv; w3  [o + 2] = r2 * inv;
    }
}

// ---------------------------------------------------------------------------
// Kernel 2a/2b: build inverse map pixel -> point index (or -1).
// ---------------------------------------------------------------------------
__global__ void init_inv_kernel(int* __restrict__ inv)
{
    const int i = (int)(blockIdx.x * blockDim.x + threadIdx.x);
    if (i < NPIX) inv[i] = -1;
}

__global__ void scatter_inv_kernel(const int* __restrict__ nz,
                                   int* __restrict__ inv, int N)
{
    const int i = (int)(blockIdx.x * blockDim.x + threadIdx.x);
    if (i < N) {
        const int q = nz[i];
        if (q >= 0 && q < NPIX) inv[q] = i;
    }
}

// ---------------------------------------------------------------------------
// Kernel 3: fused 3-NN interpolate + ksz x ksz average pool.
// One block per output pixel (b, ho, wo); one thread per feature channel.
// Window metadata (idx, w) cached in LDS; feature gathers are coalesced and
// hit L2 (feature table is only ~3 MB vs 192 MB L2).
// ---------------------------------------------------------------------------
__global__ void pool_interp_kernel(const float* __restrict__ feat, // (B,G,Dim)
                                   const int*   __restrict__ inv,  // (NPIX)
                                   const int*   __restrict__ idx3, // (B,N,3)
                                   const float* __restrict__ w3,   // (B,N,3)
                                   float* __restrict__ out,        // (B,Dim,Ho,Ho)
                                   int B, int N, int G, int Dim, int ksz, int Ho)
{
    __shared__ int   sIdx[MAXWIN * 3];
    __shared__ float sW  [MAXWIN * 3];
    __shared__ int   sCount;

    const int blk = (int)blockIdx.x;
    const int wo  = blk % Ho;
    const int t1  = blk / Ho;
    const int ho  = t1 % Ho;
    const int b   = t1 / Ho;
    const int d   = (int)threadIdx.x;

    const int win = ksz * ksz;
    if (d == 0) sCount = 0;
    __syncthreads();

    if (d < win && d < MAXWIN) {
        const int py = ho * ksz + d / ksz;
        const int px = wo * ksz + d % ksz;
        const int n  = inv[py * IMG + px];
        if (n >= 0) {
            const int slot = atomicAdd(&sCount, 1);
            const size_t o = ((size_t)b * N + n) * 3;
            sIdx[slot * 3 + 0] = idx3[o + 0];
            sIdx[slot * 3 + 1] = idx3[o + 1];
            sIdx[slot * 3 + 2] = idx3[o + 2];
            sW  [slot * 3 + 0] = w3  [o + 0];
            sW  [slot * 3 + 1] = w3  [o + 1];
            sW  [slot * 3 + 2] = w3  [o + 2];
        }
    }
    __syncthreads();

    const int cnt = sCount;
    const float* fb = feat + (size_t)b * G * Dim;
    float acc = 0.0f;
    for (int j = 0; j < cnt; ++j) {
        acc += sW[j * 3 + 0] * fb[(size_t)sIdx[j * 3 + 0] * Dim + d]
             + sW[j * 3 + 1] * fb[(size_t)sIdx[j * 3 + 1] * Dim + d]
             + sW[j * 3 + 2] * fb[(size_t)sIdx[j * 3 + 2] * Dim + d];
    }
    out[(((size_t)b * Dim + d) * Ho + ho) * Ho + wo] = acc / (float)win;
}

// ---------------------------------------------------------------------------
extern "C" void kernel_launch(void* const* d_in, const int* in_sizes, int n_in,
                              void* d_out, int out_size, void* d_ws, size_t ws_size,
                              hipStream_t stream)
{
    const float* feat = (const float*)d_in[0];   // (B,G,Dim)
    const float* src  = (const float*)d_in[1];   // (B,G,3)
    const float* tgt  = (const float*)d_in[2];   // (B,N,3)
    const int*   nz   = (const int*)  d_in[3];   // (N)
    float*       out  = (float*)d_out;           // (B,Dim,Ho,Ho)

    const int N   = in_sizes[3];
    const int B   = in_sizes[2] / (3 * N);
    const int G   = in_sizes[1] / (3 * B);
    const int Dim = in_sizes[0] / (B * G);
    const int hh  = out_size / (B * Dim);        // Ho*Ho
    int Ho = 1;
    while ((Ho + 1) * (Ho + 1) <= hh) ++Ho;      // integer sqrt
    const int ksz = IMG / Ho;                    // 8 for the reference setup

    // Workspace partition (needs ~2.6 MB)
    int*   idx3 = (int*)d_ws;
    float* w3   = (float*)(idx3 + (size_t)B * N * 3);
    int*   inv  = (int*)(w3 + (size_t)B * N * 3);

    const int    tiles  = B * ((N + 15) / 16);
    const size_t shmem1 = (size_t)(16 * G + 16) * sizeof(float);
    knn_weights_kernel<<<tiles, 32, shmem1, stream>>>(tgt, src, idx3, w3, B, N, G);

    init_inv_kernel<<<(NPIX + 255) / 256, 256, 0, stream>>>(inv);
    scatter_inv_kernel<<<(N + 255) / 256, 256, 0, stream>>>(nz, inv, N);

    const int blocks = B * Ho * Ho;
    pool_interp_kernel<<<blocks, Dim, 0, stream>>>(feat, inv, idx3, w3, out,
                                                   B, N, G, Dim, ksz, Ho);
}